// EdgeSoftmax_81655918232271
// MI455X (gfx1250) — compile-verified
//
#include <hip/hip_runtime.h>
#include <climits>

// ---------------------------------------------------------------------------
// Edge softmax (scatter-max / exp / scatter-sum), H = 8 heads per edge.
//   in[0]: logits f32 [E, H, 1]   (contiguous: 8 floats = 32B per edge)
//   in[1]: dst    i32 [E]
//   in[2]: num_nodes (scalar, unused on device; N derived from out_size)
//   out  : scores f32 [E, H] ++ normalizer f32 [N, H]
//   ws   : int32 max-table [N, H] (order-preserving float<->int mapping)
// ---------------------------------------------------------------------------

#define EH_HEADS 8

typedef float v4f __attribute__((ext_vector_type(4)));
typedef int   v4i __attribute__((ext_vector_type(4)));

// Order-preserving bijection f32 -> i32 (monotonic for all finite floats and inf).
__device__ __forceinline__ int f32_to_ord(float x) {
    int b = __float_as_int(x);
    return (b >= 0) ? b : (b ^ 0x7FFFFFFF);
}
__device__ __forceinline__ float ord_to_f32(int b) {
    return __int_as_float((b >= 0) ? b : (b ^ 0x7FFFFFFF));
}

// Re-arm max table to sentinel (< ord(-inf)) and zero the normalizer.
// Must run every kernel_launch call: atomics below accumulate otherwise.
__global__ void es_init_kernel(int* __restrict__ wsmax,
                               float* __restrict__ norm,
                               int nh) {
    int i = blockIdx.x * blockDim.x + threadIdx.x;
    if (i < nh) {
        wsmax[i] = INT_MIN;   // below ord(-inf) = 0x807FFFFF
        norm[i]  = 0.0f;
    }
}

// Pass 1: per-edge vectorized load (2 x b128) + 8 hardware i32 atomic-max into
// the 3.2 MB node table (L2-resident).
__global__ void es_max_kernel(const float* __restrict__ logits,
                              const int* __restrict__ dst,
                              int* __restrict__ wsmax,
                              int E) {
    int e = blockIdx.x * blockDim.x + threadIdx.x;
    if (e >= E) return;

    const int d = dst[e];
    const v4f* lp = reinterpret_cast<const v4f*>(logits + (size_t)e * EH_HEADS);
    v4f a = lp[0];
    v4f b = lp[1];

    int* base = wsmax + (size_t)d * EH_HEADS;
    atomicMax(base + 0, f32_to_ord(a.x));
    atomicMax(base + 1, f32_to_ord(a.y));
    atomicMax(base + 2, f32_to_ord(a.z));
    atomicMax(base + 3, f32_to_ord(a.w));
    atomicMax(base + 4, f32_to_ord(b.x));
    atomicMax(base + 5, f32_to_ord(b.y));
    atomicMax(base + 6, f32_to_ord(b.z));
    atomicMax(base + 7, f32_to_ord(b.w));
}

// Pass 2: scores = exp(logit - max[dst]) streamed out non-temporally;
// normalizer accumulated with hardware global_atomic_add_f32 (L2-resident).
__global__ void es_score_kernel(const float* __restrict__ logits,
                                const int* __restrict__ dst,
                                const int* __restrict__ wsmax,
                                float* __restrict__ scores,
                                float* __restrict__ norm,
                                int E) {
    int e = blockIdx.x * blockDim.x + threadIdx.x;
    if (e >= E) return;

    const int d = dst[e];
    const v4f* lp = reinterpret_cast<const v4f*>(logits + (size_t)e * EH_HEADS);
    v4f a = lp[0];
    v4f b = lp[1];

    const v4i* mp = reinterpret_cast<const v4i*>(wsmax + (size_t)d * EH_HEADS);
    v4i m0 = mp[0];
    v4i m1 = mp[1];

    v4f s0, s1;
    s0.x = __expf(a.x - ord_to_f32(m0.x));
    s0.y = __expf(a.y - ord_to_f32(m0.y));
    s0.z = __expf(a.z - ord_to_f32(m0.z));
    s0.w = __expf(a.w - ord_to_f32(m0.w));
    s1.x = __expf(b.x - ord_to_f32(m1.x));
    s1.y = __expf(b.y - ord_to_f32(m1.y));
    s1.z = __expf(b.z - ord_to_f32(m1.z));
    s1.w = __expf(b.w - ord_to_f32(m1.w));

    // Streamed output, never re-read: bypass near caches.
    v4f* sp = reinterpret_cast<v4f*>(scores + (size_t)e * EH_HEADS);
    __builtin_nontemporal_store(s0, sp + 0);
    __builtin_nontemporal_store(s1, sp + 1);

    float* nbase = norm + (size_t)d * EH_HEADS;
    atomicAdd(nbase + 0, s0.x);
    atomicAdd(nbase + 1, s0.y);
    atomicAdd(nbase + 2, s0.z);
    atomicAdd(nbase + 3, s0.w);
    atomicAdd(nbase + 4, s1.x);
    atomicAdd(nbase + 5, s1.y);
    atomicAdd(nbase + 6, s1.z);
    atomicAdd(nbase + 7, s1.w);
}

extern "C" void kernel_launch(void* const* d_in, const int* in_sizes, int n_in,
                              void* d_out, int out_size, void* d_ws, size_t ws_size,
                              hipStream_t stream) {
    const float* logits = (const float*)d_in[0];
    const int*   dst    = (const int*)d_in[1];

    const int E  = in_sizes[1];              // 3,200,000
    const int EH = in_sizes[0];              // E * H
    const int NH = out_size - EH;            // N * H (normalizer elements)

    float* scores = (float*)d_out;
    float* norm   = (float*)d_out + (size_t)EH;
    int*   wsmax  = (int*)d_ws;              // NH ints (3.2 MB)

    const int BLK = 256;                     // 8 wave32 waves per block

    // Pass 0: re-init accumulators (required every call; graph replays reuse state).
    es_init_kernel<<<(NH + BLK - 1) / BLK, BLK, 0, stream>>>(wsmax, norm, NH);

    // Pass 1: scatter-max.
    es_max_kernel<<<(E + BLK - 1) / BLK, BLK, 0, stream>>>(logits, dst, wsmax, E);

    // Pass 2: exp + scatter-sum.
    es_score_kernel<<<(E + BLK - 1) / BLK, BLK, 0, stream>>>(logits, dst, wsmax,
                                                             scores, norm, E);
}